// MultiHeadGAT_32169305047302
// MI455X (gfx1250) — compile-verified
//
#include <hip/hip_runtime.h>
#include <math.h>

#define BB   8
#define NN   1024
#define FIN  128
#define FOUT 64
#define HH   4
#define EPSF 1e-7f
#define NEG  0.2f

typedef float v2f __attribute__((ext_vector_type(2)));
typedef float v8f __attribute__((ext_vector_type(8)));

// D = A(16x4 f32) * B(4x16 f32) + C(16x16 f32) on one wave32.
__device__ __forceinline__ v8f wmma_f32(v2f a, v2f b, v8f c) {
    return __builtin_amdgcn_wmma_f32_16x16x4_f32(
        /*neg_a=*/false, a, /*neg_b=*/false, b,
        /*c_mod=*/(short)0, c, /*reuse_a=*/false, /*reuse_b=*/false);
}

// ---------------------------------------------------------------------------
// Kernel 1: Wh[b,h,n,o] = sum_f x[b,n,f] * W[h,f,o]
// One wave per (b, h, 16-row block). 4 o-tiles of 16, K=128 in steps of 4.
// ---------------------------------------------------------------------------
__global__ __launch_bounds__(32) void gat_wh(const float* __restrict__ x,
                                             const float* __restrict__ W,
                                             float* __restrict__ Wh) {
    const int lane = threadIdx.x;
    const int lm = lane & 15, hi = lane >> 4;
    int bid = blockIdx.x;
    const int ib = bid & 63; bid >>= 6;   // N/16 = 64
    const int h  = bid & 3;  bid >>= 2;   // H = 4
    const int b  = bid;
    const int i0 = ib * 16;

    // A-fragment source: row M = lm, K pair at k + 2*hi
    const float* xr = x + ((size_t)b * NN + i0 + lm) * FIN + 2 * hi;
    // B-fragment source: col N = lm (+16*t), K rows k + 2*hi, k + 2*hi + 1
    const float* Wp = W + (size_t)h * FIN * FOUT + (size_t)(2 * hi) * FOUT + lm;

    v8f acc0 = {}, acc1 = {}, acc2 = {}, acc3 = {};
    #pragma unroll 4
    for (int k = 0; k < FIN; k += 4) {
        v2f a = *(const v2f*)(xr + k);
        const float* wk = Wp + (size_t)k * FOUT;
        v2f b0 = { wk[0],  wk[FOUT]      };
        v2f b1 = { wk[16], wk[FOUT + 16] };
        v2f b2 = { wk[32], wk[FOUT + 32] };
        v2f b3 = { wk[48], wk[FOUT + 48] };
        acc0 = wmma_f32(a, b0, acc0);
        acc1 = wmma_f32(a, b1, acc1);
        acc2 = wmma_f32(a, b2, acc2);
        acc3 = wmma_f32(a, b3, acc3);
    }
    // C layout: VGPR r -> rows r (lanes 0-15) / r+8 (lanes 16-31), col = lm
    float* outp = Wh + (((size_t)b * HH + h) * NN + i0 + 8 * hi) * FOUT + lm;
    #pragma unroll
    for (int r = 0; r < 8; ++r) {
        float* o = outp + (size_t)r * FOUT;
        o[0] = acc0[r]; o[16] = acc1[r]; o[32] = acc2[r]; o[48] = acc3[r];
    }
}

// ---------------------------------------------------------------------------
// Kernel 2: s1[bhn] = Wh[bhn,:] . a1[h],  s2[bhn] = Wh[bhn,:] . a2[h]
// ---------------------------------------------------------------------------
__global__ void gat_scores(const float* __restrict__ Wh,
                           const float* __restrict__ a_w,
                           float* __restrict__ s1, float* __restrict__ s2) {
    int idx = blockIdx.x * blockDim.x + threadIdx.x;   // bh*N + n
    if (idx >= BB * HH * NN) return;
    const int h = (idx / NN) & (HH - 1);
    const float* row = Wh + (size_t)idx * FOUT;
    const float* a1 = a_w + (size_t)h * 2 * FOUT;
    const float* a2 = a1 + FOUT;
    float v1 = 0.f, v2 = 0.f;
    #pragma unroll 8
    for (int o = 0; o < FOUT; ++o) {
        float w = row[o];
        v1 += w * a1[o];
        v2 += w * a2[o];
    }
    s1[idx] = v1; s2[idx] = v2;
}

// ---------------------------------------------------------------------------
// Kernel 3: D[b,h,j] = sum_i A[b,i,j] * exp(lrelu(s1[i]+s2[j]+ab[h]))
// Thread per (b,h,j); A reads coalesced across j.
// ---------------------------------------------------------------------------
__global__ void gat_colsum(const float* __restrict__ A,
                           const float* __restrict__ s1,
                           const float* __restrict__ s2,
                           const float* __restrict__ a_b,
                           float* __restrict__ D) {
    int bid = blockIdx.x;
    const int jblk = bid & 3; bid >>= 2;  // N/256 = 4
    const int h    = bid & 3; bid >>= 2;
    const int b    = bid;
    const int j  = jblk * 256 + threadIdx.x;
    const int bh = b * HH + h;

    const float s2v = s2[(size_t)bh * NN + j];
    const float ab  = a_b[h];
    const float* Ac  = A + (size_t)b * NN * NN + j;
    const float* s1p = s1 + (size_t)bh * NN;

    float acc = 0.f;
    for (int i = 0; i < NN; ++i) {
        float e = s1p[i] + s2v + ab;
        e = (e >= 0.f) ? e : NEG * e;
        acc += Ac[(size_t)i * NN] * expf(e);
    }
    D[(size_t)bh * NN + j] = acc;
}

// ---------------------------------------------------------------------------
// Kernel 4: fold column normalization into Wh:  Wh[bhj,:] *= 1/(D[bhj]+eps)
// ---------------------------------------------------------------------------
__global__ void gat_scale(float* __restrict__ Wh, const float* __restrict__ D) {
    size_t idx = (size_t)blockIdx.x * blockDim.x + threadIdx.x;
    if (idx >= (size_t)BB * HH * NN * FOUT) return;
    Wh[idx] *= 1.0f / (D[idx >> 6] + EPSF);
}

// ---------------------------------------------------------------------------
// Kernel 5: out[b,i, h*64+o] = sum_j E[b,h,i,j] * Whs[b,h,j,o]
// One wave per (b,h,16-row block). Builds a 16x64 E tile in LDS (each exp
// computed once), then 16 K-steps x 4 o-tiles of f32 WMMA per tile.
// ---------------------------------------------------------------------------
__global__ __launch_bounds__(32) void gat_attn_out(
        const float* __restrict__ A, const float* __restrict__ Whs,
        const float* __restrict__ s1, const float* __restrict__ s2,
        const float* __restrict__ a_b, float* __restrict__ out) {
    __shared__ __align__(16) float E[16 * 68];   // stride 68 -> conflict-free b64

    const int lane = threadIdx.x;
    const int lm = lane & 15, hi = lane >> 4;
    int bid = blockIdx.x;
    const int ib = bid & 63; bid >>= 6;
    const int h  = bid & 3;  bid >>= 2;
    const int b  = bid;
    const int i0 = ib * 16;
    const int bh = b * HH + h;

    const float ab = a_b[h];
    const float* s1p = s1 + (size_t)bh * NN + i0;
    const float* s2p = s2 + (size_t)bh * NN;
    const float* Ap  = A  + (size_t)b * NN * NN;
    const float* Wp  = Whs + (size_t)bh * NN * FOUT;

    v8f acc0 = {}, acc1 = {}, acc2 = {}, acc3 = {};

    for (int j0 = 0; j0 < NN; j0 += 64) {
        const float s2a = s2p[j0 + lane];
        const float s2b = s2p[j0 + lane + 32];
        // Build E tile: 16 rows x 64 cols, 2 elements per lane per row.
        #pragma unroll 4
        for (int r = 0; r < 16; ++r) {
            const float s1v = s1p[r];
            const float* Ar = Ap + (size_t)(i0 + r) * NN + j0;
            float e0 = s1v + s2a + ab; e0 = (e0 >= 0.f) ? e0 : NEG * e0;
            float e1 = s1v + s2b + ab; e1 = (e1 >= 0.f) ? e1 : NEG * e1;
            E[r * 68 + lane]      = Ar[lane]      * expf(e0);
            E[r * 68 + lane + 32] = Ar[lane + 32] * expf(e1);
        }
        __syncthreads();
        #pragma unroll
        for (int kk = 0; kk < 16; ++kk) {
            v2f a = *(const v2f*)&E[lm * 68 + kk * 4 + 2 * hi];
            const float* wr = Wp + (size_t)(j0 + kk * 4 + 2 * hi) * FOUT + lm;
            v2f b0 = { wr[0],  wr[FOUT]      };
            v2f b1 = { wr[16], wr[FOUT + 16] };
            v2f b2 = { wr[32], wr[FOUT + 32] };
            v2f b3 = { wr[48], wr[FOUT + 48] };
            acc0 = wmma_f32(a, b0, acc0);
            acc1 = wmma_f32(a, b1, acc1);
            acc2 = wmma_f32(a, b2, acc2);
            acc3 = wmma_f32(a, b3, acc3);
        }
        __syncthreads();
    }

    // out[b, i, h*FOUT + o] : transpose(0,2,1,3).reshape fold
    float* op = out + ((size_t)b * NN + i0 + 8 * hi) * (HH * FOUT) + (size_t)h * FOUT + lm;
    #pragma unroll
    for (int r = 0; r < 8; ++r) {
        float* o = op + (size_t)r * (HH * FOUT);
        o[0] = acc0[r]; o[16] = acc1[r]; o[32] = acc2[r]; o[48] = acc3[r];
    }
}

// ---------------------------------------------------------------------------
extern "C" void kernel_launch(void* const* d_in, const int* in_sizes, int n_in,
                              void* d_out, int out_size, void* d_ws, size_t ws_size,
                              hipStream_t stream) {
    const float* A  = (const float*)d_in[0];   // (B,N,N)
    const float* x  = (const float*)d_in[1];   // (B,N,F_IN)
    const float* W  = (const float*)d_in[2];   // (H,F_IN,F_OUT)
    const float* aw = (const float*)d_in[3];   // (H,2*F_OUT)
    const float* ab = (const float*)d_in[4];   // (H,)
    float* out = (float*)d_out;                // (B,N,H*F_OUT)

    // Workspace layout (floats): Wh | s1 | s2 | D   (~8.8 MB total)
    float* Wh = (float*)d_ws;
    float* s1 = Wh + (size_t)BB * HH * NN * FOUT;
    float* s2 = s1 + (size_t)BB * HH * NN;
    float* D  = s2 + (size_t)BB * HH * NN;

    gat_wh      <<<dim3(BB * HH * (NN / 16)),  dim3(32),  0, stream>>>(x, W, Wh);
    gat_scores  <<<dim3(BB * HH * NN / 256),   dim3(256), 0, stream>>>(Wh, aw, s1, s2);
    gat_colsum  <<<dim3(BB * HH * (NN / 256)), dim3(256), 0, stream>>>(A, s1, s2, ab, D);
    gat_scale   <<<dim3((BB * HH * NN * FOUT) / 256), dim3(256), 0, stream>>>(Wh, D);
    gat_attn_out<<<dim3(BB * HH * (NN / 16)),  dim3(32),  0, stream>>>(A, Wh, s1, s2, ab, out);
}